// TriangleRasterizerIntegrandSlang_5222680232017
// MI455X (gfx1250) — compile-verified
//
#include <hip/hip_runtime.h>
#include <hip/hip_bf16.h>
#include <cstdint>

#define GRIDN   256
#define KMAX    64
#define RESN    512
#define NCELLS  (GRIDN * GRIDN)
#define CELLS_PER_BLK 128          // raster block covers 128 contiguous cells

typedef __attribute__((ext_vector_type(2))) float v2f;
typedef __attribute__((ext_vector_type(8))) float v8f;
typedef __attribute__((ext_vector_type(4))) unsigned int v4u;
typedef __attribute__((ext_vector_type(4))) int v4i;
typedef __attribute__((ext_vector_type(8))) int v8i;

// ---- monotonic float <-> uint encoding for atomic min/max ----
__device__ __forceinline__ unsigned fenc(float f) {
    unsigned b = __float_as_uint(f);
    return (b & 0x80000000u) ? ~b : (b | 0x80000000u);
}
__device__ __forceinline__ float fdec(unsigned e) {
    unsigned b = (e & 0x80000000u) ? (e ^ 0x80000000u) : ~e;
    return __uint_as_float(b);
}

// stats layout (floats/uints at d_ws base):
// [0..2] column sums  [3..5] min enc  [6..8] max enc  [9..11] mean  [12] scale

__global__ void init_stats(float* stats) {
    unsigned* su = (unsigned*)stats;
    stats[0] = 0.f; stats[1] = 0.f; stats[2] = 0.f;
    su[3] = 0xFFFFFFFFu; su[4] = 0xFFFFFFFFu; su[5] = 0xFFFFFFFFu;
    su[6] = 0u; su[7] = 0u; su[8] = 0u;
}

__global__ void stats_kernel(const float* __restrict__ u, int nv, float* stats) {
    __shared__ float lsum[3];
    __shared__ unsigned lmn[3], lmx[3];
    if (threadIdx.x < 3) {
        lsum[threadIdx.x] = 0.f;
        lmn[threadIdx.x] = 0xFFFFFFFFu;
        lmx[threadIdx.x] = 0u;
    }
    __syncthreads();

    float s0 = 0.f, s1 = 0.f, s2 = 0.f;
    float mn0 = 3.4e38f, mn1 = 3.4e38f, mn2 = 3.4e38f;
    float mx0 = -3.4e38f, mx1 = -3.4e38f, mx2 = -3.4e38f;
    int stride = gridDim.x * blockDim.x;
    for (int i = blockIdx.x * blockDim.x + threadIdx.x; i < nv; i += stride) {
        float a = u[i * 3 + 0], b = u[i * 3 + 1], c = u[i * 3 + 2];
        s0 += a; s1 += b; s2 += c;
        mn0 = fminf(mn0, a); mn1 = fminf(mn1, b); mn2 = fminf(mn2, c);
        mx0 = fmaxf(mx0, a); mx1 = fmaxf(mx1, b); mx2 = fmaxf(mx2, c);
    }
    atomicAdd(&lsum[0], s0); atomicAdd(&lsum[1], s1); atomicAdd(&lsum[2], s2);
    atomicMin(&lmn[0], fenc(mn0)); atomicMin(&lmn[1], fenc(mn1)); atomicMin(&lmn[2], fenc(mn2));
    atomicMax(&lmx[0], fenc(mx0)); atomicMax(&lmx[1], fenc(mx1)); atomicMax(&lmx[2], fenc(mx2));
    __syncthreads();
    if (threadIdx.x < 3) {
        unsigned* su = (unsigned*)stats;
        atomicAdd(&stats[threadIdx.x], lsum[threadIdx.x]);
        atomicMin(&su[3 + threadIdx.x], lmn[threadIdx.x]);
        atomicMax(&su[6 + threadIdx.x], lmx[threadIdx.x]);
    }
}

__global__ void finalize_stats(float* stats, int nv) {
    unsigned* su = (unsigned*)stats;
    float vmax = -3.4e38f, vmin = 3.4e38f;
    float inv = 1.0f / (float)nv;
    for (int c = 0; c < 3; ++c) {
        float mean = stats[c] * inv;
        stats[9 + c] = mean;
        vmax = fmaxf(vmax, fdec(su[6 + c]) - mean);
        vmin = fminf(vmin, fdec(su[3 + c]) - mean);
    }
    float span = fmaxf(vmax - vmin, 1e-6f);
    stats[12] = 0.5f / span;
}

// ---- vertex transform via V_WMMA_F32_16X16X4_F32 ----
// D(16x16) = A(16x4) x B(4x16); A rows = 16 vertices' homogeneous coords,
// B[k][n] = M[n][k] (columns n>=4 zero / ignored).
__global__ void vtx_wmma(const float* __restrict__ u,
                         const float* __restrict__ mvp,   // mvps[0], row-major 4x4
                         const float* __restrict__ stats,
                         float* __restrict__ vndc,
                         int nv) {
    int wave = threadIdx.x >> 5;
    int lane = threadIdx.x & 31;
    int group = blockIdx.x * (blockDim.x >> 5) + wave;
    int base = group * 16;
    if (base >= nv) return;            // wave-uniform

    float m0 = stats[9], m1 = stats[10], m2 = stats[11], s = stats[12];
    int n = lane & 15;
    int vid = base + n;
    if (vid > nv - 1) vid = nv - 1;
    float vx = (u[vid * 3 + 0] - m0) * s;
    float vy = (u[vid * 3 + 1] - m1) * s;
    float vz = (u[vid * 3 + 2] - m2) * s;

    bool lo = (lane < 16);
    v2f a;
    a.x = lo ? vx : vz;
    a.y = lo ? vy : 1.0f;

    // branchless B: load mvp row min(n,3) as float4, mask columns n>=4
    int nc = (n < 3) ? n : 3;
    float4 mr = ((const float4*)mvp)[nc];
    float msk = (n < 4) ? 1.0f : 0.0f;
    v2f b;
    b.x = (lo ? mr.x : mr.z) * msk;    // K0 / K2 rows
    b.y = (lo ? mr.y : mr.w) * msk;    // K1 / K3 rows

    v8f c = {};
    c = __builtin_amdgcn_wmma_f32_16x16x4_f32(
        /*neg_a=*/false, a, /*neg_b=*/false, b,
        /*c_mod=*/(short)0, c, /*reuse_a=*/false, /*reuse_b=*/false);

    int mb = lo ? 0 : 8;
    bool full = (base + 16 <= nv);     // wave-uniform
    if (n < 4) {
        float* dst = vndc + (size_t)(base + mb) * 4 + n;
        if (full) {
#pragma unroll
            for (int r = 0; r < 8; ++r) dst[r * 4] = c[r];
        } else {
#pragma unroll
            for (int r = 0; r < 8; ++r)
                if (base + mb + r < nv) dst[r * 4] = c[r];
        }
    }
}

// ---- per-triangle setup + binning ----
__global__ void tri_setup(const float* __restrict__ u,
                          const int* __restrict__ faces,
                          const float* __restrict__ vndc,
                          const float* __restrict__ light,
                          const float* __restrict__ base_colors,
                          const float* __restrict__ stats,
                          float* __restrict__ tridata,
                          int* __restrict__ cnt,
                          int* __restrict__ grid,
                          int F) {
    int f = blockIdx.x * blockDim.x + threadIdx.x;
    if (f >= F) return;

    int i0 = faces[f * 3 + 0], i1 = faces[f * 3 + 1], i2 = faces[f * 3 + 2];
    float c0x = vndc[i0 * 4 + 0], c0y = vndc[i0 * 4 + 1], c0z = vndc[i0 * 4 + 2];
    float c1x = vndc[i1 * 4 + 0], c1y = vndc[i1 * 4 + 1], c1z = vndc[i1 * 4 + 2];
    float c2x = vndc[i2 * 4 + 0], c2y = vndc[i2 * 4 + 1], c2z = vndc[i2 * 4 + 2];

    float p0x = (c0x + 1.f) * 0.5f, p0y = (c0y + 1.f) * 0.5f;
    float p1x = (c1x + 1.f) * 0.5f, p1y = (c1y + 1.f) * 0.5f;
    float p2x = (c2x + 1.f) * 0.5f, p2y = (c2y + 1.f) * 0.5f;
    float z = (c0z + c1z + c2z) * (1.0f / 3.0f);

    float m0 = stats[9], m1 = stats[10], m2 = stats[11], s = stats[12];
    float w0x = (u[i0*3+0]-m0)*s, w0y = (u[i0*3+1]-m1)*s, w0z = (u[i0*3+2]-m2)*s;
    float w1x = (u[i1*3+0]-m0)*s, w1y = (u[i1*3+1]-m1)*s, w1z = (u[i1*3+2]-m2)*s;
    float w2x = (u[i2*3+0]-m0)*s, w2y = (u[i2*3+1]-m1)*s, w2z = (u[i2*3+2]-m2)*s;
    float e1x = w0x - w1x, e1y = w0y - w1y, e1z = w0z - w1z;
    float e2x = w2x - w1x, e2y = w2y - w1y, e2z = w2z - w1z;
    float nx = e1y * e2z - e1z * e2y;
    float ny = e1z * e2x - e1x * e2z;
    float nz = e1x * e2y - e1y * e2x;
    float invl = 1.0f / (sqrtf(nx*nx + ny*ny + nz*nz) + 1e-10f);
    nx *= invl; ny *= invl; nz *= invl;
    float sh = fmaxf(0.f, -(light[0]*nx + light[1]*ny + light[2]*nz));
    float cr = base_colors[f*3+0] * sh;
    float cg = base_colors[f*3+1] * sh;
    float cb = base_colors[f*3+2] * sh;

    float4* q = (float4*)(tridata + (size_t)f * 12);
    q[0] = make_float4(p0x, p0y, p1x, p1y);
    q[1] = make_float4(p2x, p2y, z, cr);
    q[2] = make_float4(cg, cb, 0.f, 0.f);

    float mnx = fminf(p0x, fminf(p1x, p2x));
    float mny = fminf(p0y, fminf(p1y, p2y));
    float mxx = fmaxf(p0x, fmaxf(p1x, p2x));
    float mxy = fmaxf(p0y, fmaxf(p1y, p2y));
    int lox = min(max((int)floorf(mnx * GRIDN), 0), GRIDN - 1);
    int loy = min(max((int)floorf(mny * GRIDN), 0), GRIDN - 1);
    int hix = min(max((int)floorf(mxx * GRIDN), 0), GRIDN - 1);
    int hiy = min(max((int)floorf(mxy * GRIDN), 0), GRIDN - 1);

#pragma unroll
    for (int dy = 0; dy < 4; ++dy) {
        int cy = loy + dy;
        if (cy > hiy) break;
#pragma unroll
        for (int dx = 0; dx < 4; ++dx) {
            int cx = lox + dx;
            if (cx > hix) break;
            int cell = cy * GRIDN + cx;
            int idx = atomicAdd(&cnt[cell], 1);
            if (idx < KMAX) grid[(size_t)cell * KMAX + idx] = f;
        }
    }
}

// ---- per-pixel raster ----
// Each 256-thread block covers half a pixel row: y fixed, x in [xb, xb+256),
// i.e. 128 contiguous grid cells.  Their candidate lists (128*64 ints = 32KB,
// contiguous in `grid`) are staged into LDS with one Tensor-Data-Mover load
// issued by wave 0, then every pixel scans its cell's list from LDS.
__global__ void raster(const float* __restrict__ tridata,
                       const int* __restrict__ cnt,
                       const int* __restrict__ grid,
                       const float* __restrict__ bg,
                       float* __restrict__ out) {
    __shared__ int slist[CELLS_PER_BLK * KMAX];   // 32 KB

    int t = threadIdx.x;
    int blk = blockIdx.x;
    int xb = (blk & 1) << 8;          // 0 or 256
    int y  = blk >> 1;
    int x  = xb + t;
    int iy = y >> 1;
    int cxb = xb >> 1;                // first cell x for this block
    int cellbase = iy * GRIDN + cxb;
    const int* src = grid + (size_t)cellbase * KMAX;   // 8192 contiguous ints

    __builtin_prefetch(cnt + cellbase, 0, 0);          // global_prefetch_b8

#if __has_builtin(__builtin_amdgcn_tensor_load_to_lds)
    if (t < 32) {                     // one wave issues the DMA (TDM ignores EXEC)
        const unsigned NW = CELLS_PER_BLK * KMAX;      // 8192 dwords
        unsigned lds_addr = (unsigned)(uintptr_t)&slist[0];
        unsigned long long ga = (unsigned long long)(uintptr_t)src;
        v4u g0;
        g0.x = 1u;                                           // count=1, user D#
        g0.y = lds_addr;                                     // lds_addr
        g0.z = (unsigned)(ga & 0xFFFFFFFFu);                 // global_addr[31:0]
        g0.w = (unsigned)((ga >> 32) & 0x01FFFFFFu) | (2u << 30); // addr[56:32] | type=2
        v8i g1;
        g1[0] = (int)(2u << 16);                             // data_size = 4B
        g1[1] = (int)((NW & 0xFFFFu) << 16);                 // tensor_dim0[15:0]
        g1[2] = (int)(((NW >> 16) & 0xFFFFu) | (1u << 16));  // tensor_dim0 hi | tensor_dim1=1
        g1[3] = (int)((NW & 0xFFFFu) << 16);                 // tile_dim0 = NW
        g1[4] = 0;                                           // tile_dim1/2 unused
        g1[5] = (int)NW;                                     // tensor_dim0_stride
        g1[6] = 0;
        g1[7] = 0;
        v4i gz = {0, 0, 0, 0};
# if __clang_major__ >= 23
        v8i gz8 = {0, 0, 0, 0, 0, 0, 0, 0};
        __builtin_amdgcn_tensor_load_to_lds(g0, g1, gz, gz, gz8, 0);
# else
        __builtin_amdgcn_tensor_load_to_lds(g0, g1, gz, gz, 0);
# endif
        __builtin_amdgcn_s_wait_tensorcnt(0);
    }
#else
    for (int i = t; i < CELLS_PER_BLK * KMAX; i += blockDim.x) slist[i] = src[i];
#endif
    __syncthreads();

    int ci = t >> 1;                  // local cell index, 0..127
    int cell = cellbase + ci;
    const int* lst = &slist[ci * KMAX];

    float pu = (x + 0.5f) * (1.0f / RESN);
    float pv = (y + 0.5f) * (1.0f / RESN);

    int n = cnt[cell];
    if (n > KMAX) n = KMAX;

    float bz = 3.4e38f;
    float r = bg[0], g = bg[1], b = bg[2];
    for (int k = 0; k < n; ++k) {
        int tri = lst[k];
        const float4* q = (const float4*)(tridata + (size_t)tri * 12);
        float4 q0 = q[0];
        float4 q1 = q[1];
        float ax = q0.x, ay = q0.y, bx = q0.z, by = q0.w;
        float cx = q1.x, cy = q1.y, z = q1.z;
        float e0 = (bx - ax) * (pv - ay) - (by - ay) * (pu - ax);
        float e1 = (cx - bx) * (pv - by) - (cy - by) * (pu - bx);
        float e2 = (ax - cx) * (pv - cy) - (ay - cy) * (pu - cx);
        bool inside = (e0 >= 0.f && e1 >= 0.f && e2 >= 0.f) ||
                      (e0 <= 0.f && e1 <= 0.f && e2 <= 0.f);
        if (inside && z < bz) {
            bz = z;
            float4 q2 = q[2];
            r = q1.w; g = q2.x; b = q2.y;
        }
    }
    size_t id = (size_t)y * RESN + x;
    out[id * 3 + 0] = r;
    out[id * 3 + 1] = g;
    out[id * 3 + 2] = b;
}

extern "C" void kernel_launch(void* const* d_in, const int* in_sizes, int n_in,
                              void* d_out, int out_size, void* d_ws, size_t ws_size,
                              hipStream_t stream) {
    const float* u           = (const float*)d_in[0];
    const float* mvps        = (const float*)d_in[1];
    const float* light       = (const float*)d_in[2];
    const float* bg          = (const float*)d_in[3];
    const float* base_colors = (const float*)d_in[4];
    const int*   faces       = (const int*)d_in[5];
    float* out = (float*)d_out;

    int NV = in_sizes[0] / 3;
    int F  = in_sizes[5] / 3;

    char* ws = (char*)d_ws;
    float* stats = (float*)ws;                                  // 64 B used
    int* cnt     = (int*)(ws + 256);                            // NCELLS * 4
    size_t off   = 256 + (size_t)NCELLS * sizeof(int);
    off = (off + 255) & ~(size_t)255;
    float* vndc  = (float*)(ws + off);                          // NV * 4 floats
    off += (size_t)NV * 4 * sizeof(float);
    off = (off + 255) & ~(size_t)255;
    float* tridata = (float*)(ws + off);                        // F * 12 floats
    off += (size_t)F * 12 * sizeof(float);
    off = (off + 255) & ~(size_t)255;
    int* grid = (int*)(ws + off);                               // NCELLS * KMAX ints

    init_stats<<<1, 1, 0, stream>>>(stats);
    hipMemsetAsync(cnt, 0, (size_t)NCELLS * sizeof(int), stream);

    stats_kernel<<<256, 256, 0, stream>>>(u, NV, stats);
    finalize_stats<<<1, 1, 0, stream>>>(stats, NV);

    int ngroups = (NV + 15) / 16;          // 16 vertices per wave
    int nblocksB = (ngroups + 7) / 8;      // 8 waves per 256-thread block
    vtx_wmma<<<nblocksB, 256, 0, stream>>>(u, mvps, stats, vndc, NV);

    tri_setup<<<(F + 255) / 256, 256, 0, stream>>>(u, faces, vndc, light,
                                                   base_colors, stats,
                                                   tridata, cnt, grid, F);

    raster<<<(RESN * RESN) / 256, 256, 0, stream>>>(tridata, cnt, grid, bg, out);
}